// FlattenConv_11656541241415
// MI455X (gfx1250) — compile-verified
//
#include <hip/hip_runtime.h>
#include <stddef.h>

// ---------------------------------------------------------------------------
// Patch-unfold (depthwise conv with diagonal-identity 3x3 kernel).
//   out[n, c*9 + ky*3+kx, h, w] = x[n, c, h-1+ky, w-1+kx]   (zero padded)
// Pure data movement: ~51 MB read (L2-resident), ~462 MB streamed write.
// Bandwidth floor on MI455X (23.3 TB/s): ~22 us. No FLOPs -> no WMMA.
// CDNA5 paths used: async global->LDS (ASYNCcnt), s_wait_asynccnt,
// non-temporal b128 stores, wave32-sized 256-thread (8-wave) blocks.
// ---------------------------------------------------------------------------

#define NB   16
#define C_CH 64
#define H    112
#define W    112
#define RPB  8              // output rows per block
#define HB   (H / RPB)      // 14 row-blocks
#define W4C  (W / 4)        // 28 float4 columns per row
#define LROWS (RPB + 2)     // 10 LDS rows (with top/bottom halo)
#define LSTR  120           // LDS row stride in floats; data at cols [4..115]
                            // col 3 = left zero halo, col 116 = right zero halo
                            // (4+w0)*4 and LSTR*4 are both 16B multiples -> aligned b128

typedef float v4f __attribute__((ext_vector_type(4)));
typedef int   v4i __attribute__((ext_vector_type(4)));
typedef __attribute__((address_space(1))) v4i as1_v4i;   // global ("__device__")
typedef __attribute__((address_space(3))) v4i as3_v4i;   // LDS    ("__shared__")

#if defined(__has_builtin)
#if __has_builtin(__builtin_amdgcn_global_load_async_to_lds_b128) && \
    __has_builtin(__builtin_amdgcn_s_wait_asynccnt)
#define USE_ASYNC_LDS 1
#endif
#endif

__global__ __launch_bounds__(256)
void FlattenConv_unfold_kernel(const float* __restrict__ x,
                               float* __restrict__ out) {
    __shared__ float lds[LROWS][LSTR];

    const int tid = threadIdx.x;
    int b = blockIdx.x;
    const int hb = b % HB;   b /= HB;
    const int c  = b % C_CH; b /= C_CH;
    const int n  = b;
    const int h0 = hb * RPB;

    const float* __restrict__ xplane =
        x + (size_t)(n * C_CH + c) * (size_t)(H * W);

    // ---- Stage 10 input rows (h0-1 .. h0+8) into LDS, zero out-of-range ----
    for (int task = tid; task < LROWS * W4C; task += 256) {
        const int r  = task / W4C;       // LDS row
        const int w4 = task % W4C;
        const int g  = h0 - 1 + r;       // global input row
        float* dst = &lds[r][4 + 4 * w4];
        if ((unsigned)g < (unsigned)H) {
            const float* src = xplane + (size_t)g * W + 4 * w4;
#ifdef USE_ASYNC_LDS
            __builtin_amdgcn_global_load_async_to_lds_b128(
                (as1_v4i*)src, (as3_v4i*)dst, 0, 0);
#else
            *(v4f*)dst = *(const v4f*)src;
#endif
        } else {
            v4f z = {0.f, 0.f, 0.f, 0.f};
            *(v4f*)dst = z;
        }
        if (w4 == 0) {                   // zero halo columns (branch-free reads)
            lds[r][3]     = 0.f;
            lds[r][4 + W] = 0.f;
        }
    }
#ifdef USE_ASYNC_LDS
    __builtin_amdgcn_s_wait_asynccnt(0);
#endif
    __syncthreads();

    // ---- Each of 224 threads owns one (h, w0..w0+3) slab; 9 NT b128 stores ----
    if (tid < RPB * W4C) {
        const int hl = tid / W4C;
        const int w4 = tid % W4C;
        const int w0 = 4 * w4;
        const int h  = h0 + hl;

        const size_t plane = (size_t)(H * W);
        const size_t ob = (size_t)((n * C_CH + c) * 9) * plane
                        + (size_t)h * W + (size_t)w0;

#pragma unroll
        for (int dy = 0; dy < 3; ++dy) {
            const int r = hl + dy;       // LDS row holding input row h-1+dy
            v4f   a  = *(const v4f*)&lds[r][4 + w0];  // cols w0..w0+3
            float lf = lds[r][3 + w0];                // col  w0-1 (or halo 0)
            float rf = lds[r][8 + w0];                // col  w0+4 (or halo 0)

            v4f v0 = {lf,  a.x, a.y, a.z};           // kx=0 (shift -1)
            v4f v1 = a;                               // kx=1 (center)
            v4f v2 = {a.y, a.z, a.w, rf};            // kx=2 (shift +1)

            __builtin_nontemporal_store(v0, (v4f*)(out + ob + (size_t)(dy * 3 + 0) * plane));
            __builtin_nontemporal_store(v1, (v4f*)(out + ob + (size_t)(dy * 3 + 1) * plane));
            __builtin_nontemporal_store(v2, (v4f*)(out + ob + (size_t)(dy * 3 + 2) * plane));
        }
    }
}

extern "C" void kernel_launch(void* const* d_in, const int* in_sizes, int n_in,
                              void* d_out, int out_size, void* d_ws, size_t ws_size,
                              hipStream_t stream) {
    (void)in_sizes; (void)n_in; (void)d_ws; (void)ws_size; (void)out_size;
    const float* x = (const float*)d_in[0];   // weight (d_in[1]) is identity by
    float* out = (float*)d_out;               // construction -> unused
    const int blocks = NB * C_CH * HB;        // 16*64*14 = 14336
    FlattenConv_unfold_kernel<<<blocks, 256, 0, stream>>>(x, out);
}